// UpsampleConvexWithMask_3882650437064
// MI455X (gfx1250) — compile-verified
//
#include <hip/hip_runtime.h>
#include <hip/hip_bf16.h>
#include <math.h>

// UpsampleConvexWithMask (RAFT convex upsampling), SCALE=8
// x:    (4, 120, 160, 2)   float32
// mask: (4, 120, 160, 576) float32   [576 = 9 * 64, index = k*64 + sc]
// out:  (4, 960, 1280, 2)  float32
//
// Memory-bound kernel (~216 MB moved, ~0.25 GFLOP): stream the mask through
// LDS with CDNA5 async global->LDS copies, do softmax + 9-tap weighted sum in
// VALU, write coalesced float2 outputs.

#define B_  4
#define H_  120
#define W_  160
#define C_  2
#define K_  9
#define SS_ 64          // s*s = 8*8
#define MPP 576         // mask floats per pixel (K_*SS_)
#define PIX_PER_BLOCK 4

__global__ __launch_bounds__(256)
void upsample_convex_kernel(const float* __restrict__ x,
                            const float* __restrict__ mask,
                            float* __restrict__ out)
{
    __shared__ float smask[PIX_PER_BLOCK * MPP];   // 9216 B
    __shared__ float sx[PIX_PER_BLOCK][K_][C_];    //  288 B

    const int tid   = threadIdx.x;          // 0..255
    const int group = tid >> 6;             // pixel within block, 0..3
    const int sc    = tid & 63;             // sub-position 0..63

    const int pix = blockIdx.x * PIX_PER_BLOCK + group;   // < 76800 exactly
    const int b   = pix / (H_ * W_);
    const int hw  = pix % (H_ * W_);
    const int h   = hw / W_;
    const int w   = hw % W_;

    // ---- async copy of this block's mask tile: global -> LDS, b128 lanes ----
    {
        const float* mbase = mask + (size_t)blockIdx.x * (PIX_PER_BLOCK * MPP);
        // 4*576 floats = 576 float4 transfers, 256 threads -> up to 3 each
        #pragma unroll
        for (int it = 0; it < 3; ++it) {
            int idx = tid + it * 256;
            if (idx < (PIX_PER_BLOCK * MPP) / 4) {
                unsigned lds_addr = (unsigned)(size_t)(&smask[idx * 4]);
                unsigned voff     = (unsigned)(idx * 16);   // byte offset
                asm volatile(
                    "global_load_async_to_lds_b128 %0, %1, %2"
                    :
                    : "v"(lds_addr), "v"(voff), "s"(mbase)
                    : "memory");
            }
        }
    }

    // ---- stage 3x3 neighborhood of x (zero-padded) into LDS ----
    if (sc < K_) {
        const int di = sc / 3 - 1;
        const int dj = sc % 3 - 1;
        const int hh = h + di;
        const int ww = w + dj;
        float v0 = 0.0f, v1 = 0.0f;
        if (hh >= 0 && hh < H_ && ww >= 0 && ww < W_) {
            const float* p = x + ((size_t)(b * H_ + hh) * W_ + ww) * C_;
            v0 = p[0];
            v1 = p[1];
        }
        sx[group][sc][0] = v0;
        sx[group][sc][1] = v1;
    }

    // wait for async tile, then make LDS visible to all 8 waves
    asm volatile("s_wait_asynccnt 0x0" ::: "memory");
    __syncthreads();

    // ---- per-thread: 9-way softmax (over k) and weighted sum ----
    const float* mp = &smask[group * MPP + sc];   // stride 64 floats: bank-conflict-free
    float mv[K_];
    #pragma unroll
    for (int k = 0; k < K_; ++k) mv[k] = mp[k * SS_];

    float mx = mv[0];
    #pragma unroll
    for (int k = 1; k < K_; ++k) mx = fmaxf(mx, mv[k]);

    float sum = 0.0f, acc0 = 0.0f, acc1 = 0.0f;
    #pragma unroll
    for (int k = 0; k < K_; ++k) {
        const float e = __expf(mv[k] - mx);   // v_exp_f32
        sum  += e;
        acc0 += e * sx[group][k][0];
        acc1 += e * sx[group][k][1];
    }
    const float inv = 1.0f / sum;

    // ---- scatter into 8x8 output tile: sc = i*8 + j ----
    const int i = sc >> 3;
    const int j = sc & 7;
    const size_t row = (size_t)(b * (H_ * 8) + h * 8 + i);
    const size_t o   = (row * (W_ * 8) + (size_t)(w * 8 + j)) * C_;
    float2 r;
    r.x = acc0 * inv;
    r.y = acc1 * inv;
    *(float2*)(out + o) = r;   // coalesced global_store_b64
}

extern "C" void kernel_launch(void* const* d_in, const int* in_sizes, int n_in,
                              void* d_out, int out_size, void* d_ws, size_t ws_size,
                              hipStream_t stream)
{
    const float* x    = (const float*)d_in[0];   // 4*120*160*2
    const float* mask = (const float*)d_in[1];   // 4*120*160*576
    float* out        = (float*)d_out;           // 4*960*1280*2

    const int n_pix   = B_ * H_ * W_;            // 76800
    const int n_block = n_pix / PIX_PER_BLOCK;   // 19200, exact

    upsample_convex_kernel<<<dim3(n_block), dim3(256), 0, stream>>>(x, mask, out);
}